// GraphSAGEv2_5007931867342
// MI455X (gfx1250) — compile-verified
//
#include <hip/hip_runtime.h>

typedef __attribute__((ext_vector_type(2))) float v2f;
typedef __attribute__((ext_vector_type(8))) float v8f;

#define DD 128      // feature dim
#define KC 32       // K-chunk per LDS stage
#define SSTR 36     // padded LDS row stride (floats): conflict-free b64 frag reads
#define TM 128      // rows per block (8 waves x 16)
#define SCAN_E 1024 // elements scanned per block (256 thr x 4)

// ---------------- degree count (int atomics, once per launch) ----------------
__global__ void sage_count_kernel(const int* __restrict__ dst,
                                  int* __restrict__ cnt, int E) {
    int e = blockIdx.x * blockDim.x + threadIdx.x;
    if (e < E) atomicAdd(&cnt[dst[e]], 1);
}

__global__ void sage_inv_kernel(const int* __restrict__ cnt,
                                float* __restrict__ inv, int n) {
    int i = blockIdx.x * blockDim.x + threadIdx.x;
    if (i < n) inv[i] = 1.0f / ((float)cnt[i] + 1.0f);
}

// ---------------- two-level exclusive scan over cnt -> rowStart ----------------
__global__ __launch_bounds__(256)
void scan_block_kernel(const int* __restrict__ in, int* __restrict__ out,
                       int* __restrict__ blockSums, int n) {
    __shared__ int sh[256];
    int t = threadIdx.x;
    int base = blockIdx.x * SCAN_E + t * 4;
    int v[4]; int s = 0;
#pragma unroll
    for (int k = 0; k < 4; ++k) {
        int idx = base + k;
        int x = (idx < n) ? in[idx] : 0;
        v[k] = s;            // exclusive within thread
        s += x;
    }
    sh[t] = s;
    __syncthreads();
    for (int off = 1; off < 256; off <<= 1) {
        int val = (t >= off) ? sh[t - off] : 0;
        __syncthreads();
        sh[t] += val;
        __syncthreads();
    }
    int tbase = (t > 0) ? sh[t - 1] : 0;
#pragma unroll
    for (int k = 0; k < 4; ++k) {
        int idx = base + k;
        if (idx < n) out[idx] = tbase + v[k];
    }
    if (t == 255) blockSums[blockIdx.x] = sh[255];
}

__global__ __launch_bounds__(1024)
void scan_sums_kernel(int* __restrict__ sums, int nb) {
    __shared__ int sh[1024];
    int t = threadIdx.x;
    sh[t] = (t < nb) ? sums[t] : 0;
    __syncthreads();
    for (int off = 1; off < 1024; off <<= 1) {
        int val = (t >= off) ? sh[t - off] : 0;
        __syncthreads();
        sh[t] += val;
        __syncthreads();
    }
    if (t < nb) sums[t] = (t > 0) ? sh[t - 1] : 0;
}

__global__ void scan_add_kernel(int* __restrict__ rowStart, int* __restrict__ cur,
                                const int* __restrict__ sums, int n) {
    int idx = blockIdx.x * blockDim.x + threadIdx.x;
    if (idx < n) {
        int v = rowStart[idx] + sums[idx / SCAN_E];
        rowStart[idx] = v;
        cur[idx] = v;
    }
}

// ---------------- bucket edges by dst: elist[rowStart[d]...] = src ----------------
__global__ void sage_build_csr_kernel(const int* __restrict__ src,
                                      const int* __restrict__ dst,
                                      int* __restrict__ cur,
                                      int* __restrict__ elist, int E) {
    int e = blockIdx.x * blockDim.x + threadIdx.x;
    if (e < E) {
        int p = atomicAdd(&cur[dst[e]], 1);
        elist[p] = src[e];
    }
}

// ---------------- per-layer aggregation: pure gather, no atomics ----------------
// one wave per node; lane holds 4 columns as float4
__global__ __launch_bounds__(256)
void sage_aggregate_kernel(const float* __restrict__ x,
                           const int* __restrict__ rowStart,
                           const int* __restrict__ cnt,
                           const int* __restrict__ elist,
                           float* __restrict__ agg, int n) {
    int wave = threadIdx.x >> 5;
    int lane = threadIdx.x & 31;
    int node = blockIdx.x * 8 + wave;
    if (node >= n) return;
    int start = rowStart[node];
    int deg   = cnt[node];
    const float4* __restrict__ x4 = (const float4*)x;
    float4 acc = make_float4(0.f, 0.f, 0.f, 0.f);
    for (int j = 0; j < deg; ++j) {
        int s = elist[start + j];                 // wave-uniform broadcast load
        float4 v = x4[(size_t)s * 32 + lane];     // coalesced 512B per wave
        acc.x += v.x; acc.y += v.y; acc.z += v.z; acc.w += v.w;
    }
    ((float4*)agg)[(size_t)node * 32 + lane] = acc;
}

// ---------------- fused concat-GEMM + bias + ReLU (fp32 WMMA) ----------------
// out[r, :] = relu( [x[r,:], agg[r,:]*inv[r]] @ W + b ),  W is (256,128) row-major
__global__ __launch_bounds__(256)
void sage_gemm_relu_kernel(const float* __restrict__ X, const float* __restrict__ AGG,
                           const float* __restrict__ INV, const float* __restrict__ W,
                           const float* __restrict__ Bias, float* __restrict__ OUT,
                           int n) {
    __shared__ float sA[TM * SSTR];   // 128 rows x 32 k (padded)   ~18KB
    __shared__ float sBt[DD * SSTR];  // 128 cols x 32 k (W^T tile) ~18KB

    const int tid  = threadIdx.x;
    const int lane = tid & 31;
    const int w    = tid >> 5;     // wave id 0..7 -> row sub-tile
    const int half = lane >> 4;    // 0: K pair {0,1}, 1: K pair {2,3}
    const int l    = lane & 15;    // M (for A) / N (for B,C,D)
    const int rowBase = blockIdx.x * TM;

    v8f acc[8];
#pragma unroll
    for (int j = 0; j < 8; ++j) acc[j] = (v8f)0.0f;

    for (int c = 0; c < 8; ++c) {
        const int k0 = c * KC;                     // column in concat space [0,256)
        const bool isAgg = (c >= 4);
        const float* __restrict__ srcp = isAgg ? (AGG + (size_t)(k0 - DD))
                                               : (X   + (size_t)k0);
        // stage A tile: 128 rows x 32 cols (apply inv scaling for neighbor half)
#pragma unroll
        for (int it = 0; it < (TM * KC) / 256; ++it) {
            int i  = tid + it * 256;
            int r  = i >> 5, kk = i & 31;
            int row = rowBase + r;
            float v = 0.0f;
            if (row < n) {
                v = srcp[(size_t)row * DD + kk];
                if (isAgg) v *= INV[row];
            }
            sA[r * SSTR + kk] = v;
        }
        // stage W tile transposed: sBt[col][kk] = W[k0+kk][col]
#pragma unroll
        for (int it = 0; it < (KC * DD) / 256; ++it) {
            int i  = tid + it * 256;
            int kk = i >> 7, j = i & 127;
            sBt[j * SSTR + kk] = W[(size_t)(k0 + kk) * DD + j];
        }
        __syncthreads();

#pragma unroll
        for (int kk = 0; kk < KC; kk += 4) {
            // A frag 16x4: lanes 0-15 hold K={kk,kk+1}, lanes 16-31 K={kk+2,kk+3}
            v2f a = *(const v2f*)&sA[(w * 16 + l) * SSTR + kk + half * 2];
#pragma unroll
            for (int j = 0; j < 8; ++j) {
                // B frag 4x16 (mirror layout), read from transposed W tile
                v2f b = *(const v2f*)&sBt[(j * 16 + l) * SSTR + kk + half * 2];
                acc[j] = __builtin_amdgcn_wmma_f32_16x16x4_f32(
                    false, a, false, b, (short)0, acc[j], false, false);
            }
        }
        __syncthreads();
    }

    // epilogue: C/D layout -> lane 0-15: N=lane, vgpr v: M=v; lanes 16-31: M=v+8
#pragma unroll
    for (int j = 0; j < 8; ++j) {
        int col  = j * 16 + l;
        float bj = Bias[col];
#pragma unroll
        for (int v = 0; v < 8; ++v) {
            int row = rowBase + w * 16 + half * 8 + v;
            if (row < n) {
                float val = acc[j][v] + bj;
                OUT[(size_t)row * DD + col] = val > 0.0f ? val : 0.0f;
            }
        }
    }
}

extern "C" void kernel_launch(void* const* d_in, const int* in_sizes, int n_in,
                              void* d_out, int out_size, void* d_ws, size_t ws_size,
                              hipStream_t stream) {
    const float* x  = (const float*)d_in[0];
    const int*   ei = (const int*)d_in[1];
    const float* W0 = (const float*)d_in[2];
    const float* b0 = (const float*)d_in[3];
    const float* W1 = (const float*)d_in[4];
    const float* b1 = (const float*)d_in[5];
    const float* W2 = (const float*)d_in[6];
    const float* b2 = (const float*)d_in[7];

    const int N = in_sizes[0] / DD;
    const int E = in_sizes[1] / 2;
    const int* src = ei;
    const int* dst = ei + E;

    // workspace layout
    float* agg       = (float*)d_ws;                 // N*128 f
    float* y         = agg + (size_t)N * DD;         // N*128 f
    float* inv       = y   + (size_t)N * DD;         // N f
    int*   cnt       = (int*)(inv + N);              // N i
    int*   rowStart  = cnt + N;                      // N i
    int*   cur       = rowStart + N;                 // N i
    int*   elist     = cur + N;                      // E i
    int*   blockSums = elist + E;                    // <=1024 i

    const int scanBlocks = (N + SCAN_E - 1) / SCAN_E;      // 98 for N=100K
    const int gemmBlocks = (N + TM - 1) / TM;
    const int aggBlocks  = (N + 7) / 8;

    // ---- CSR build (once) ----
    hipMemsetAsync(cnt, 0, (size_t)N * sizeof(int), stream);
    sage_count_kernel<<<(E + 255) / 256, 256, 0, stream>>>(dst, cnt, E);
    sage_inv_kernel<<<(N + 255) / 256, 256, 0, stream>>>(cnt, inv, N);
    scan_block_kernel<<<scanBlocks, 256, 0, stream>>>(cnt, rowStart, blockSums, N);
    scan_sums_kernel<<<1, 1024, 0, stream>>>(blockSums, scanBlocks);
    scan_add_kernel<<<(N + 255) / 256, 256, 0, stream>>>(rowStart, cur, blockSums, N);
    sage_build_csr_kernel<<<(E + 255) / 256, 256, 0, stream>>>(src, dst, cur, elist, E);

    // ---- layer 0: x -> y ----
    sage_aggregate_kernel<<<aggBlocks, 256, 0, stream>>>(x, rowStart, cnt, elist, agg, N);
    sage_gemm_relu_kernel<<<gemmBlocks, 256, 0, stream>>>(x, agg, inv, W0, b0, y, N);

    // ---- layer 1: y -> y (in-place safe: each row read/written by its own block) ----
    sage_aggregate_kernel<<<aggBlocks, 256, 0, stream>>>(y, rowStart, cnt, elist, agg, N);
    sage_gemm_relu_kernel<<<gemmBlocks, 256, 0, stream>>>(y, agg, inv, W1, b1, y, N);

    // ---- layer 2: y -> out ----
    sage_aggregate_kernel<<<aggBlocks, 256, 0, stream>>>(y, rowStart, cnt, elist, agg, N);
    sage_gemm_relu_kernel<<<gemmBlocks, 256, 0, stream>>>(y, agg, inv, W2, b2,
                                                          (float*)d_out, N);
}